// PDFSampler_87385404604910
// MI455X (gfx1250) — compile-verified
//
#include <hip/hip_runtime.h>

// PDF inverse-CDF sampler for MI455X (gfx1250, wave32).
// One wave = one ray. Bin edges staged to LDS via global_load_async_to_lds_b128
// (ASYNCcnt), weights scanned in registers with a wave32 shuffle scan, CDF kept
// in LDS for the per-sample binary search.

#define NUM_RAYS     131072
#define NUM_EXISTING 256
#define NUM_BINS     97          // num_samples + 1
#define WAVES_PER_BLOCK 8
#define LDS_STRIDE   272         // padded 257 (keeps 16B alignment, spreads banks)

__device__ __forceinline__ unsigned lds_byte_offset(const void* p) {
    // generic -> LDS(addrspace 3) pointer -> 32-bit byte offset within LDS
    return (unsigned)(unsigned long long)
        (__attribute__((address_space(3))) const void*)p;
}

__global__ __launch_bounds__(256, 1)
void pdf_sample_kernel(const float* __restrict__ W,    // [R, 256] weights
                       const float* __restrict__ SS,   // [R, 256] spacing_starts
                       const float* __restrict__ SE,   // [R, 256] spacing_ends
                       float* __restrict__ OUT)        // [2, R, 97]
{
    __shared__ __align__(16) float smem[WAVES_PER_BLOCK * 2 * LDS_STRIDE];

    const int lane = threadIdx.x & 31;
    const int wv   = threadIdx.x >> 5;
    const int ray  = (int)blockIdx.x * WAVES_PER_BLOCK + wv;

    float* binsS = smem + wv * 2 * LDS_STRIDE;   // 257 bin edges
    float* cdfS  = binsS + LDS_STRIDE;           // 257 CDF entries

    // ---- Phase 1: async-stage this ray's 256 bin edges (1 KB) into LDS ----
    {
        unsigned l0 = lds_byte_offset(binsS) + (unsigned)lane * 16u;
        unsigned l1 = l0 + 512u;
        unsigned g0 = (unsigned)ray * 1024u + (unsigned)lane * 16u;
        unsigned g1 = g0 + 512u;
        unsigned long long base = (unsigned long long)SS;
        asm volatile(
            "global_load_async_to_lds_b128 %0, %1, %4\n\t"
            "global_load_async_to_lds_b128 %2, %3, %4"
            :: "v"(l0), "v"(g0), "v"(l1), "v"(g1), "s"(base)
            : "memory");
    }

    // last edge comes from spacing_ends[ray, 255]
    const float lastEdge = SE[(size_t)ray * NUM_EXISTING + (NUM_EXISTING - 1)];

    // ---- Phase 2: weights -> CDF (register scan, wave32 shuffle scan) ----
    const float4* wvec =
        reinterpret_cast<const float4*>(W + (size_t)ray * NUM_EXISTING + lane * 8);
    float4 a = wvec[0];
    float4 b = wvec[1];

    float p[8];
    p[0] = a.x;        p[1] = p[0] + a.y; p[2] = p[1] + a.z; p[3] = p[2] + a.w;
    p[4] = p[3] + b.x; p[5] = p[4] + b.y; p[6] = p[5] + b.z; p[7] = p[6] + b.w;

    float lsum = p[7];
    float incl = lsum;
    #pragma unroll
    for (int d = 1; d < 32; d <<= 1) {
        float v = __shfl_up(incl, d, 32);
        if (lane >= d) incl += v;
    }
    const float excl  = incl - lsum;
    const float total = __shfl(incl, 31, 32);

    const float pad  = fmaxf(1e-5f - total, 0.0f);
    const float padc = pad * (1.0f / (float)NUM_EXISTING);
    const float inv  = 1.0f / (total + pad);

    // cdf[k] = min(1, (prefix_k + k*padc) * inv), k = 1..256 ; cdf[0] = 0
    #pragma unroll
    for (int j = 0; j < 8; ++j) {
        int   k = lane * 8 + j + 1;
        float c = (excl + p[j] + (float)k * padc) * inv;
        cdfS[k] = fminf(c, 1.0f);
    }

    // drain this wave's async copies before anyone reads binsS
    asm volatile("s_wait_asynccnt 0" ::: "memory");
    if (lane == 0) {
        cdfS[0] = 0.0f;
        binsS[NUM_EXISTING] = lastEdge;
    }
    __syncthreads();

    // ---- Phase 3: 97 midpoint samples, right-bisection + lerp ----
    float* outE = OUT + (size_t)ray * NUM_BINS;                               // euclidean
    float* outS = OUT + (size_t)NUM_RAYS * NUM_BINS + (size_t)ray * NUM_BINS; // spacing

    const float invNB = 1.0f / (float)NUM_BINS;
    #pragma unroll
    for (int kk = 0; kk < 4; ++kk) {
        int j = lane + kk * 32;
        if (j < NUM_BINS) {
            float u = ((float)j + 0.5f) * invNB;

            // searchsorted(cdf, u, side='right') over 257 entries
            int lo = 0, hi = NUM_EXISTING + 1;
            while (lo < hi) {
                int mid = (lo + hi) >> 1;
                if (u < cdfS[mid]) hi = mid; else lo = mid + 1;
            }
            int below = lo - 1;
            if (below < 0) below = 0;
            if (below > NUM_EXISTING) below = NUM_EXISTING;
            int above = lo;
            if (above > NUM_EXISTING) above = NUM_EXISTING;

            float c0 = cdfS[below], c1 = cdfS[above];
            float b0 = binsS[below], b1 = binsS[above];
            float den = c1 - c0;
            if (den < 1e-5f) den = 1.0f;
            float t = (u - c0) / den;
            t = fminf(fmaxf(t, 0.0f), 1.0f);
            float bin = b0 + t * (b1 - b0);

            outE[j] = 2.0f + bin * 4.0f;   // NEAR + bin * (FAR - NEAR)
            outS[j] = bin;
        }
    }
}

extern "C" void kernel_launch(void* const* d_in, const int* in_sizes, int n_in,
                              void* d_out, int out_size, void* d_ws, size_t ws_size,
                              hipStream_t stream) {
    (void)in_sizes; (void)n_in; (void)out_size; (void)d_ws; (void)ws_size;
    const float* weights = (const float*)d_in[0];
    const float* sstarts = (const float*)d_in[1];
    const float* sends   = (const float*)d_in[2];
    float* out = (float*)d_out;

    dim3 grid(NUM_RAYS / WAVES_PER_BLOCK);   // 16384 blocks
    dim3 block(256);                          // 8 wave32 waves
    pdf_sample_kernel<<<grid, block, 0, stream>>>(weights, sstarts, sends, out);
}